// GCNModel_51831665328310
// MI455X (gfx1250) — compile-verified
//
#include <hip/hip_runtime.h>
#include <hip/hip_bf16.h>

typedef float v2f __attribute__((ext_vector_type(2)));
typedef float v8f __attribute__((ext_vector_type(8)));

#define N_NODES 100000
#define N_EDGES 1600000
#define SCAN_B 256

// ---------------- degree / normalization ----------------
__global__ void k_count(const long long* __restrict__ ei, int* __restrict__ cnt, int E) {
    int i = blockIdx.x * blockDim.x + threadIdx.x;
    if (i < E) atomicAdd(&cnt[(int)ei[(long long)E + i]], 1);
}

__global__ void k_dinv(const int* __restrict__ cnt, float* __restrict__ dinv, int n) {
    int i = blockIdx.x * blockDim.x + threadIdx.x;
    if (i < n) dinv[i] = rsqrtf((float)(cnt[i] + 1));   // +1 self-loop; always > 0
}

// ---------------- exclusive scan (3-phase) ----------------
__global__ void k_scan_block(const int* __restrict__ in, int* __restrict__ out,
                             int* __restrict__ bsums, int n) {
    __shared__ int s[SCAN_B];
    int tid = threadIdx.x;
    int i = blockIdx.x * SCAN_B + tid;
    int v = (i < n) ? in[i] : 0;
    s[tid] = v;
    __syncthreads();
    for (int off = 1; off < SCAN_B; off <<= 1) {
        int t = (tid >= off) ? s[tid - off] : 0;
        __syncthreads();
        s[tid] += t;
        __syncthreads();
    }
    if (i < n) out[i] = s[tid] - v;                 // exclusive
    if (tid == SCAN_B - 1) bsums[blockIdx.x] = s[tid];
}

__global__ void k_scan_sums(int* __restrict__ bsums, int nb) {  // one block of 512
    __shared__ int s[512];
    int tid = threadIdx.x;
    int v = (tid < nb) ? bsums[tid] : 0;
    s[tid] = v;
    __syncthreads();
    for (int off = 1; off < 512; off <<= 1) {
        int t = (tid >= off) ? s[tid - off] : 0;
        __syncthreads();
        s[tid] += t;
        __syncthreads();
    }
    if (tid < nb) bsums[tid] = s[tid] - v;          // exclusive
}

__global__ void k_scan_add(int* __restrict__ out, const int* __restrict__ bsums, int n) {
    int i = blockIdx.x * SCAN_B + threadIdx.x;
    if (i < n) out[i] += bsums[blockIdx.x];
}

// ---------------- CSR fill (dst-major) ----------------
__global__ void k_fill(const long long* __restrict__ ei, const int* __restrict__ rowptr,
                       int* __restrict__ fill, const float* __restrict__ dinv,
                       int* __restrict__ csr_src, float* __restrict__ csr_w, int E) {
    int i = blockIdx.x * blockDim.x + threadIdx.x;
    if (i >= E) return;
    int s = (int)ei[i];
    int d = (int)ei[(long long)E + i];
    int pos = rowptr[d] + atomicAdd(&fill[d], 1);
    csr_src[pos] = s;
    csr_w[pos]   = dinv[s] * dinv[d];               // constant across all 3 layers
}

// ---------------- fp32 WMMA GEMM 128->128: Y = X[Nx128] @ W[128x128] ----------------
// grid.x = N/16 (one 16-row stripe per block); 8 waves = 8 column tiles.
// X tile staged in LDS (row stride 132 floats: conflict-free 16-lane column reads,
// 132*4 bytes is 16B-aligned for float4 staging stores).
__global__ __launch_bounds__(256) void k_gemm128(const float* __restrict__ X,
                                                 const float* __restrict__ W,
                                                 float* __restrict__ Y) {
    __shared__ float lx[16 * 132];
    int tm  = blockIdx.x;
    int tid = threadIdx.x;

    // stage X[16][128] -> lx, coalesced float4 (512 float4, 2 per thread)
    {
        const float4* src = (const float4*)(X + (long)tm * (16 * 128));
#pragma unroll
        for (int it = 0; it < 2; it++) {
            int idx  = tid + it * 256;          // 0..511
            int row  = idx >> 5;                // 32 float4 per row
            int col4 = idx & 31;
            float4 v = src[row * 32 + col4];
            *(float4*)&lx[row * 132 + col4 * 4] = v;
        }
    }
    __syncthreads();

    int wv   = tid >> 5;                        // tn = 0..7
    int lane = tid & 31;
    int n    = (wv << 4) + (lane & 15);         // output column (always < 128)
    int kh   = (lane >> 4) << 1;                // upper half-wave: K+2,K+3
    const float* xr = &lx[(lane & 15) * 132];

    v8f acc = {};
#pragma unroll 8
    for (int k = 0; k < 128; k += 4) {
        v2f a, b;
        a.x = xr[k + kh];
        a.y = xr[k + kh + 1];
        b.x = W[(k + kh) * 128 + n];
        b.y = W[(k + kh + 1) * 128 + n];
        acc = __builtin_amdgcn_wmma_f32_16x16x4_f32(false, a, false, b,
                                                    (short)0, acc, false, false);
    }
    int rbase = (tm << 4) + ((lane >> 4) << 3); // VGPR r: M=r (lanes<16) / M=r+8
#pragma unroll
    for (int r = 0; r < 8; r++)
        Y[(long)(rbase + r) * 128 + n] = acc[r];
}

// ---------------- fp32 WMMA GEMM 128->40: W staged zero-padded to 128x48 in LDS ----------------
__global__ __launch_bounds__(256) void k_gemm40(const float* __restrict__ X,
                                                const float* __restrict__ W,
                                                float* __restrict__ Y) {
    __shared__ float lx[16 * 132];
    __shared__ float lw[128 * 48];              // zero-padded weight image (24 KB)
    int tm  = blockIdx.x;
    int tid = threadIdx.x;

    for (int idx = tid; idx < 128 * 48; idx += 256) {
        int r = idx / 48;
        int c = idx - r * 48;
        lw[idx] = (c < 40) ? W[r * 40 + c] : 0.0f;
    }
    {
        const float4* src = (const float4*)(X + (long)tm * (16 * 128));
#pragma unroll
        for (int it = 0; it < 2; it++) {
            int idx  = tid + it * 256;
            int row  = idx >> 5;
            int col4 = idx & 31;
            float4 v = src[row * 32 + col4];
            *(float4*)&lx[row * 132 + col4 * 4] = v;
        }
    }
    __syncthreads();

    int wv   = tid >> 5;
    int lane = tid & 31;
    if (wv >= 3) return;                        // wave-uniform exit; EXEC stays all-1s
    int n  = (wv << 4) + (lane & 15);           // 0..47 (padded)
    int kh = (lane >> 4) << 1;
    const float* xr = &lx[(lane & 15) * 132];

    v8f acc = {};
#pragma unroll 8
    for (int k = 0; k < 128; k += 4) {
        v2f a, b;
        a.x = xr[k + kh];
        a.y = xr[k + kh + 1];
        b.x = lw[(k + kh) * 48 + n];
        b.y = lw[(k + kh + 1) * 48 + n];
        acc = __builtin_amdgcn_wmma_f32_16x16x4_f32(false, a, false, b,
                                                    (short)0, acc, false, false);
    }
    if (n < 40) {
        int rbase = (tm << 4) + ((lane >> 4) << 3);
#pragma unroll
        for (int r = 0; r < 8; r++)
            Y[(long)(rbase + r) * 40 + n] = acc[r];
    }
}

// ---------------- aggregation F=128: one wave32 per node, float4 per lane ----------------
__global__ __launch_bounds__(256) void k_agg128(const float* __restrict__ G,
                                                const float* __restrict__ dinv,
                                                const int* __restrict__ rowptr,
                                                const int* __restrict__ cnt,
                                                const int* __restrict__ csr_src,
                                                const float* __restrict__ csr_w,
                                                const float* __restrict__ bias,
                                                float* __restrict__ out, int n) {
    int node = (blockIdx.x << 3) + (threadIdx.x >> 5);
    if (node >= n) return;
    int lane = threadIdx.x & 31;
    const float4* Grow = (const float4*)G;      // 32 float4 per node row

    float di = dinv[node];
    float sw = di * di;
    float4 g = Grow[(long)node * 32 + lane];    // self-loop term
    float4 acc;
    acc.x = sw * g.x; acc.y = sw * g.y; acc.z = sw * g.z; acc.w = sw * g.w;

    int beg = rowptr[node];
    int num = cnt[node];
    for (int e = 0; e < num; e++) {
        int   s = csr_src[beg + e];             // uniform across wave -> single request
        float w = csr_w[beg + e];
        float4 v = Grow[(long)s * 32 + lane];   // one coalesced 512B row read
        acc.x = fmaf(w, v.x, acc.x);
        acc.y = fmaf(w, v.y, acc.y);
        acc.z = fmaf(w, v.z, acc.z);
        acc.w = fmaf(w, v.w, acc.w);
    }
    float4 b4 = ((const float4*)bias)[lane];
    acc.x = fmaxf(acc.x + b4.x, 0.0f);
    acc.y = fmaxf(acc.y + b4.y, 0.0f);
    acc.z = fmaxf(acc.z + b4.z, 0.0f);
    acc.w = fmaxf(acc.w + b4.w, 0.0f);
    ((float4*)out)[(long)node * 32 + lane] = acc;
}

// ---------------- aggregation F=40: one 64-thread block per node ----------------
__global__ void k_agg40(const float* __restrict__ G, const float* __restrict__ dinv,
                        const int* __restrict__ rowptr, const int* __restrict__ cnt,
                        const int* __restrict__ csr_src, const float* __restrict__ csr_w,
                        const float* __restrict__ bias, float* __restrict__ out) {
    int node = blockIdx.x;
    int c = threadIdx.x;
    if (c >= 40) return;
    float di = dinv[node];
    float acc = di * di * G[(long)node * 40 + c];
    int beg = rowptr[node];
    int num = cnt[node];
    for (int e = 0; e < num; e++) {
        int   s = csr_src[beg + e];
        float w = csr_w[beg + e];
        acc = fmaf(w, G[(long)s * 40 + c], acc);
    }
    acc += bias[c];
    out[(long)node * 40 + c] = fmaxf(acc, 0.0f);
}

// ---------------- log_softmax over 40 logits, one wave32 per node ----------------
__global__ void k_logsoftmax40(const float* __restrict__ in, float* __restrict__ out, int n) {
    int wid  = (int)((blockIdx.x * (long)blockDim.x + threadIdx.x) >> 5);
    int lane = threadIdx.x & 31;
    if (wid >= n) return;
    const float* row = in + (long)wid * 40;
    float v0 = row[lane];
    float v1 = (lane < 8) ? row[32 + lane] : -__builtin_inff();
    float m = fmaxf(v0, v1);
    for (int o = 16; o > 0; o >>= 1) m = fmaxf(m, __shfl_xor(m, o, 32));
    float s = expf(v0 - m) + ((lane < 8) ? expf(v1 - m) : 0.0f);
    for (int o = 16; o > 0; o >>= 1) s += __shfl_xor(s, o, 32);
    float l = m + logf(s);
    out[(long)wid * 40 + lane] = v0 - l;
    if (lane < 8) out[(long)wid * 40 + 32 + lane] = v1 - l;
}

// ---------------- launcher ----------------
extern "C" void kernel_launch(void* const* d_in, const int* in_sizes, int n_in,
                              void* d_out, int out_size, void* d_ws, size_t ws_size,
                              hipStream_t stream) {
    const float*     x  = (const float*)d_in[0];
    const long long* ei = (const long long*)d_in[1];   // int64 edge_index [2, E]
    const float* W1 = (const float*)d_in[2];
    const float* b1 = (const float*)d_in[3];
    const float* W2 = (const float*)d_in[4];
    const float* b2 = (const float*)d_in[5];
    const float* W3 = (const float*)d_in[6];
    const float* b3 = (const float*)d_in[7];
    float* outp = (float*)d_out;

    const int N = N_NODES, E = N_EDGES;

    char* ws = (char*)d_ws;
    size_t off = 0;
    auto carve = [&](size_t bytes) -> char* {
        char* p = ws + off;
        off += (bytes + 255) & ~(size_t)255;
        return p;
    };
    float* G       = (float*)carve((size_t)N * 128 * 4);
    float* H1      = (float*)carve((size_t)N * 128 * 4);
    float* H2      = (float*)carve((size_t)N * 128 * 4);
    int*   cnt     = (int*)  carve((size_t)N * 4);
    int*   rowptr  = (int*)  carve((size_t)N * 4);
    int*   fillp   = (int*)  carve((size_t)N * 4);
    float* dinv    = (float*)carve((size_t)N * 4);
    int*   csr_src = (int*)  carve((size_t)E * 4);
    float* csr_w   = (float*)carve((size_t)E * 4);
    int*   bsums   = (int*)  carve(4096);

    hipMemsetAsync(cnt,   0, (size_t)N * 4, stream);
    hipMemsetAsync(fillp, 0, (size_t)N * 4, stream);

    int tE = (E + 255) / 256;
    int tN = (N + 255) / 256;
    int nb = (N + SCAN_B - 1) / SCAN_B;                 // 391 <= 512

    k_count<<<tE, 256, 0, stream>>>(ei, cnt, E);
    k_dinv<<<tN, 256, 0, stream>>>(cnt, dinv, N);
    k_scan_block<<<nb, SCAN_B, 0, stream>>>(cnt, rowptr, bsums, N);
    k_scan_sums<<<1, 512, 0, stream>>>(bsums, nb);
    k_scan_add<<<nb, SCAN_B, 0, stream>>>(rowptr, bsums, N);
    k_fill<<<tE, 256, 0, stream>>>(ei, rowptr, fillp, dinv, csr_src, csr_w, E);

    int gemmBlocks = N / 16;                            // 6250 (N is a multiple of 16)
    int aggBlocks  = (N + 7) / 8;                       // 8 nodes per 256-thread block

    // layer 1: 128 -> 128
    k_gemm128<<<gemmBlocks, 256, 0, stream>>>(x, W1, G);
    k_agg128<<<aggBlocks, 256, 0, stream>>>(G, dinv, rowptr, cnt, csr_src, csr_w, b1, H1, N);

    // layer 2: 128 -> 128
    k_gemm128<<<gemmBlocks, 256, 0, stream>>>(H1, W2, G);
    k_agg128<<<aggBlocks, 256, 0, stream>>>(G, dinv, rowptr, cnt, csr_src, csr_w, b2, H2, N);

    // layer 3: 128 -> 40, then log_softmax
    k_gemm40<<<gemmBlocks, 256, 0, stream>>>(H2, W3, G);
    k_agg40<<<N, 64, 0, stream>>>(G, dinv, rowptr, cnt, csr_src, csr_w, b3, H1);
    k_logsoftmax40<<<(N + 7) / 8, 256, 0, stream>>>(H1, outp, N);
}